// CSV_71390946394290
// MI455X (gfx1250) — compile-verified
//
#include <hip/hip_runtime.h>
#include <math.h>

typedef __attribute__((ext_vector_type(2))) float v2f;
typedef __attribute__((ext_vector_type(8))) float v8f;

#define TILE  16          // batch rows per block
#define DDIM  128         // embedding dim
#define STR   132         // padded LDS row stride (floats): 132 % 64 = 4 -> conflict-free frag loads
#define NSLOT 6           // slot 0 = pos, slots 1..5 = neg
#define NCOLS 22          // data row: 10 ctx | center | pos | 5 neg | 5 mask

__device__ __forceinline__ float clip10(float x) {
  return fminf(fmaxf(x, -10.0f), 10.0f);
}

__global__ __launch_bounds__(256) void w2v_sgns_loss_kernel(
    const int*   __restrict__ data,     // (B, 22) int
    const float* __restrict__ gW,       // (V+1, 128)
    const float* __restrict__ sW,       // (3V, 128)
    const float* __restrict__ cw,       // (10, 128)
    float*       __restrict__ out,      // [pos_loss, neg_loss]
    int B)
{
  __shared__ __align__(16) float s_ctx[TILE * STR];            //  8.4 KB
  __shared__ __align__(16) float s_emb[NSLOT * TILE * STR];    // 50.7 KB
  __shared__ float s_ips[NSLOT * TILE];
  __shared__ float s_pos, s_neg;

  const int tid     = threadIdx.x;
  const int lane    = tid & 31;
  const int wave    = tid >> 5;            // 8 waves of 32
  const int rowBase = blockIdx.x * TILE;

  if (tid == 0) { s_pos = 0.0f; s_neg = 0.0f; }

  // ---------- phase 1a: ctx_feats tile (16 rows x 128 dims) ----------
  // thread (r = tid>>4, c = tid&15) owns dims [8c, 8c+8) of row r
  {
    const int r  = tid >> 4;
    const int c  = tid & 15;
    const int gr = rowBase + r;
    float4 a0 = make_float4(0.f, 0.f, 0.f, 0.f);
    float4 a1 = make_float4(0.f, 0.f, 0.f, 0.f);
    if (gr < B) {
      const long base = (long)gr * NCOLS;
      #pragma unroll
      for (int j = 0; j < 10; ++j) {
        const int idx  = data[base + j];
        const float* g = gW + (long)idx * DDIM + c * 8;
        const float* w = cw + j * DDIM + c * 8;
        const float4 g0 = ((const float4*)g)[0];
        const float4 g1 = ((const float4*)g)[1];
        const float4 w0 = ((const float4*)w)[0];
        const float4 w1 = ((const float4*)w)[1];
        a0.x = fmaf(g0.x, w0.x, a0.x); a0.y = fmaf(g0.y, w0.y, a0.y);
        a0.z = fmaf(g0.z, w0.z, a0.z); a0.w = fmaf(g0.w, w0.w, a0.w);
        a1.x = fmaf(g1.x, w1.x, a1.x); a1.y = fmaf(g1.y, w1.y, a1.y);
        a1.z = fmaf(g1.z, w1.z, a1.z); a1.w = fmaf(g1.w, w1.w, a1.w);
      }
    }
    float* dst = s_ctx + r * STR + c * 8;
    ((float4*)dst)[0] = a0;
    ((float4*)dst)[1] = a1;
  }

  // ---------- phase 1b: gather pos + neg embeddings into LDS ----------
  // 96 rows total (6 slots x 16 tile rows); each wave loads 12 full rows,
  // one 512B row per wave-instruction (lane -> float4 of dims [4*lane, 4*lane+4))
  {
    #pragma unroll
    for (int i = 0; i < 12; ++i) {
      const int e    = wave * 12 + i;      // 0..95, slot-major
      const int slot = e >> 4;
      const int r    = e & 15;
      const int gr   = rowBase + r;
      float4 v = make_float4(0.f, 0.f, 0.f, 0.f);
      if (gr < B) {
        // slot 0 -> col 11 (pos); slot s in 1..5 -> col 11+s (neg s-1)
        const int idx = data[(long)gr * NCOLS + 11 + slot];
        v = ((const float4*)(sW + (long)idx * DDIM))[lane];
      }
      ((float4*)(s_emb + e * STR))[lane] = v;   // e*STR floats = e*528 B, 16B aligned
    }
  }

  __syncthreads();

  // ---------- phase 2: WMMA f32 16x16x4 chained over K=128 ----------
  // Wave s (< 6) computes D = CtxFeats(16x128) * Emb_s(128x16)^T; diag(D) = inner products.
  if (wave < NSLOT) {
    const int s  = wave;
    const int mr = lane & 15;            // A: row M / B: col N owned by this lane
    const int ko = (lane >> 4) * 2;      // lanes 16..31 hold K+2, K+3
    const float* aBase = s_ctx + mr * STR + ko;
    const float* bBase = s_emb + (s * TILE + mr) * STR + ko;
    v8f acc = {};
#if defined(__gfx1250__)
    #pragma unroll
    for (int k = 0; k < 32; ++k) {
      const v2f a = *(const v2f*)(aBase + 4 * k);   // ds_load_b64, bank-conflict free
      const v2f b = *(const v2f*)(bBase + 4 * k);
      acc = __builtin_amdgcn_wmma_f32_16x16x4_f32(
          /*neg_a=*/false, a, /*neg_b=*/false, b,
          /*c_mod=*/(short)0, acc, /*reuse_a=*/false, /*reuse_b=*/false);
    }
#endif
    // Diagonal extraction from C/D layout:
    //   lanes 0..15 : VGPR i = D[M=i  ][N=lane]   -> lane i    holds diag m=i   in acc[i]
    //   lanes 16..31: VGPR i = D[M=i+8][N=lane-16]-> lane 24+i holds diag m=i+8 in acc[i]
    #pragma unroll
    for (int i = 0; i < 8; ++i) {
      if (lane == i)      s_ips[s * TILE + i]     = acc[i];
      if (lane == 24 + i) s_ips[s * TILE + i + 8] = acc[i];
    }
  }

  __syncthreads();

  // ---------- phase 3: losses + block reduction ----------
  if (tid < NSLOT * TILE) {
    const int slot = tid >> 4;
    const int r    = tid & 15;
    const int gr   = rowBase + r;
    if (gr < B) {
      const float ip = s_ips[tid];
      if (slot == 0) {
        // -log_sigmoid(clip(ip)) = log1p(exp(-clip(ip)))
        atomicAdd(&s_pos, log1pf(expf(-clip10(ip))));
      } else {
        const float m = (float)data[(long)gr * NCOLS + 16 + slot]; // mask cols 17..21
        atomicAdd(&s_neg, log1pf(expf(-clip10(-ip))) * m);
      }
    }
  }

  __syncthreads();
  if (tid == 0) {
    atomicAdd(out + 0, s_pos);
    atomicAdd(out + 1, s_neg);
  }
}

extern "C" void kernel_launch(void* const* d_in, const int* in_sizes, int n_in,
                              void* d_out, int out_size, void* d_ws, size_t ws_size,
                              hipStream_t stream) {
  const int*   data = (const int*)  d_in[0];   // (B, 22)
  const float* gW   = (const float*)d_in[1];   // (V+1, 128)
  const float* sW   = (const float*)d_in[2];   // (3V, 128)
  const float* cw   = (const float*)d_in[3];   // (10, 128)
  // d_in[4] = window (5), d_in[5] = negative (5): fixed by the problem, baked in.

  const int B = in_sizes[0] / NCOLS;

  float* out = (float*)d_out;
  hipMemsetAsync(out, 0, 2 * sizeof(float), stream);   // graph-capture safe

  const int blocks = (B + TILE - 1) / TILE;            // 8192 for B=131072
  w2v_sgns_loss_kernel<<<blocks, 256, 0, stream>>>(data, gW, sW, cw, out, B);
}